// EntmaxBisect_53120155517188
// MI455X (gfx1250) — compile-verified
//
#include <hip/hip_runtime.h>

// Entmax-1.5 bisection along last dim (d = 4096), rows register-resident.
// One workgroup (256 threads = 8 wave32) per row; 16 floats/thread in VGPRs.
// Global -> LDS staging uses gfx1250 async-to-LDS (ASYNCcnt) instructions.
// Trans-bound: 2 TRANS (v_log/v_exp) + 4 VALU per element per pass; the final
// normalization pass is fused into the last bisection iteration (50 passes,
// not 51).

#define DSIZE 4096
#define BLOCK 256
#define F4_PER_ROW (DSIZE / 4)              // 1024 float4 per row (16 KB)
#define F4_PER_THREAD (F4_PER_ROW / BLOCK)  // 4 float4 per thread
#define N_ITER 50
#define P_EXP (1.0f / 4095.0f)              // 1/(d-1)
#define DIFF0 0.984375f                     // 1 - 4096^(1-1.5) = 1 - 1/64

__device__ __forceinline__ float zpow(float u_clamped) {
  // u^(1/4095) for u > 0; u == 0 -> log2 = -INF -> exp2 = 0 (exact zero
  // contribution, no cmp/cndmask needed).
  return __builtin_amdgcn_exp2f(P_EXP * __builtin_amdgcn_logf(u_clamped));
}

__global__ __launch_bounds__(BLOCK) void entmax_bisect_kernel(
    const float* __restrict__ X, float* __restrict__ Out) {
  __shared__ float4 rowbuf[F4_PER_ROW];          // 16 KB staging
  __shared__ __align__(16) float red[2][8];      // double-buffered partials

  const int tid  = threadIdx.x;
  const int wave = tid >> 5;
  const int lane = tid & 31;
  const long long row = blockIdx.x;
  const float* src = X + row * (long long)DSIZE;
  float*       dst = Out + row * (long long)DSIZE;

  // ---- Stage row global -> LDS with CDNA5 async-to-LDS (ASYNCcnt) ----
#pragma unroll
  for (int c = 0; c < F4_PER_THREAD; ++c) {
    const int f4 = c * BLOCK + tid;
    const float4* gaddr = reinterpret_cast<const float4*>(src) + f4;
    unsigned ldsoff = (unsigned)(unsigned long long)&rowbuf[f4];
    asm volatile("global_load_async_to_lds_b128 %0, %1, off"
                 :: "v"(ldsoff), "v"(gaddr) : "memory");
  }
  asm volatile("s_wait_asynccnt 0" ::: "memory");
  __syncthreads();

  // ---- LDS -> registers, pre-scale by (alpha-1) = 0.5, local max ----
  float4 xs[F4_PER_THREAD];
  float lmax = -3.402823466e38f;
#pragma unroll
  for (int c = 0; c < F4_PER_THREAD; ++c) {
    float4 v = rowbuf[c * BLOCK + tid];
    v.x *= 0.5f; v.y *= 0.5f; v.z *= 0.5f; v.w *= 0.5f;
    xs[c] = v;
    lmax = fmaxf(lmax, fmaxf(fmaxf(v.x, v.y), fmaxf(v.z, v.w)));
  }

  int ph = 0;

  // ---- workgroup max (wave butterfly + vectorized 8-partial combine) ----
#pragma unroll
  for (int off = 16; off >= 1; off >>= 1)
    lmax = fmaxf(lmax, __shfl_xor(lmax, off, 32));
  if (lane == 0) red[ph & 1][wave] = lmax;
  __syncthreads();
  {
    const float4* rb4 = reinterpret_cast<const float4*>(red[ph & 1]);
    float4 a = rb4[0], b = rb4[1];
    lmax = fmaxf(fmaxf(fmaxf(a.x, a.y), fmaxf(a.z, a.w)),
                 fmaxf(fmaxf(b.x, b.y), fmaxf(b.z, b.w)));
  }
  ph++;

  // ---- bisection: diff sequence is data-independent, only t_min tracked ----
  float t_min = lmax - 1.0f;
  float diff  = DIFF0;

  // Iterations 1..N_ITER-1 (last iteration peeled: its Z and sum ARE the
  // final output numerator/denominator).
  for (int it = 0; it < N_ITER - 1; ++it) {
    diff *= 0.5f;
    const float tt = t_min + diff;
    float s0 = 0.0f, s1 = 0.0f, s2 = 0.0f, s3 = 0.0f;
#pragma unroll
    for (int c = 0; c < F4_PER_THREAD; ++c) {
      s0 += zpow(fmaxf(xs[c].x - tt, 0.0f));
      s1 += zpow(fmaxf(xs[c].y - tt, 0.0f));
      s2 += zpow(fmaxf(xs[c].z - tt, 0.0f));
      s3 += zpow(fmaxf(xs[c].w - tt, 0.0f));
    }
    float s = (s0 + s1) + (s2 + s3);
#pragma unroll
    for (int off = 16; off >= 1; off >>= 1)
      s += __shfl_xor(s, off, 32);
    float* rb = red[ph & 1];
    if (lane == 0) rb[wave] = s;
    __syncthreads();  // single barrier per iter (double-buffered partials)
    const float4* rb4 = reinterpret_cast<const float4*>(rb);
    float4 a = rb4[0], b = rb4[1];
    const float total = ((a.x + a.y) + (a.z + a.w)) +
                        ((b.x + b.y) + (b.z + b.w));
    ph++;
    if (total >= 1.0f) t_min = tt;  // uniform across workgroup
  }

  // ---- peeled final iteration: Z kept in registers, sum reused as norm ----
  {
    diff *= 0.5f;
    const float tt = t_min + diff;
    float4 zf[F4_PER_THREAD];
    float s0 = 0.0f, s1 = 0.0f, s2 = 0.0f, s3 = 0.0f;
#pragma unroll
    for (int c = 0; c < F4_PER_THREAD; ++c) {
      float zx = zpow(fmaxf(xs[c].x - tt, 0.0f));
      float zy = zpow(fmaxf(xs[c].y - tt, 0.0f));
      float zz = zpow(fmaxf(xs[c].z - tt, 0.0f));
      float zw = zpow(fmaxf(xs[c].w - tt, 0.0f));
      zf[c].x = zx; zf[c].y = zy; zf[c].z = zz; zf[c].w = zw;
      s0 += zx; s1 += zy; s2 += zz; s3 += zw;
    }
    float s = (s0 + s1) + (s2 + s3);
#pragma unroll
    for (int off = 16; off >= 1; off >>= 1)
      s += __shfl_xor(s, off, 32);
    float* rb = red[ph & 1];
    if (lane == 0) rb[wave] = s;
    __syncthreads();
    const float4* rb4 = reinterpret_cast<const float4*>(rb);
    float4 a = rb4[0], b = rb4[1];
    const float total = ((a.x + a.y) + (a.z + a.w)) +
                        ((b.x + b.y) + (b.z + b.w));
    const float inv = 1.0f / total;
#pragma unroll
    for (int c = 0; c < F4_PER_THREAD; ++c) {
      float4 o = zf[c];
      o.x *= inv; o.y *= inv; o.z *= inv; o.w *= inv;
      reinterpret_cast<float4*>(dst)[c * BLOCK + tid] = o;
    }
  }
}

extern "C" void kernel_launch(void* const* d_in, const int* in_sizes, int n_in,
                              void* d_out, int out_size, void* d_ws, size_t ws_size,
                              hipStream_t stream) {
  (void)n_in; (void)d_ws; (void)ws_size; (void)out_size;
  const float* X = (const float*)d_in[0];
  float* Out = (float*)d_out;
  const int rows = in_sizes[0] / DSIZE;  // 8 * 2048 = 16384
  entmax_bisect_kernel<<<dim3(rows), dim3(BLOCK), 0, stream>>>(X, Out);
}